// Model_81509889343513
// MI455X (gfx1250) — compile-verified
//
#include <hip/hip_runtime.h>
#include <hip/hip_bf16.h>
#include <math.h>

#define N_NODES 100000
#define N_EDGES 1600000
#define NGRAPH  512
#define FDIM    64

typedef float v2f __attribute__((ext_vector_type(2)));
typedef float v8f __attribute__((ext_vector_type(8)));

// ---------------------------------------------------------------------------
// Kernel 1: edge scatter-add  agg[dst] += x[src]
// one thread per (edge, 4-feature chunk): float4 gather + 4 fp32 atomics
// ---------------------------------------------------------------------------
__global__ __launch_bounds__(256) void agg_kernel(const float* __restrict__ x,
                                                  const long long* __restrict__ ei,
                                                  float* __restrict__ buf) {
    int tid = blockIdx.x * 256 + threadIdx.x;
    if (tid >= N_EDGES * 16) return;
    int e = tid >> 4;
    int q = tid & 15;
    long long s = ei[e];
    long long d = ei[N_EDGES + e];
    const float4 v = *(const float4*)(x + (size_t)s * FDIM + q * 4);
    float* p = buf + (size_t)d * FDIM + q * 4;
    __hip_atomic_fetch_add(p + 0, v.x, __ATOMIC_RELAXED, __HIP_MEMORY_SCOPE_AGENT);
    __hip_atomic_fetch_add(p + 1, v.y, __ATOMIC_RELAXED, __HIP_MEMORY_SCOPE_AGENT);
    __hip_atomic_fetch_add(p + 2, v.z, __ATOMIC_RELAXED, __HIP_MEMORY_SCOPE_AGENT);
    __hip_atomic_fetch_add(p + 3, v.w, __ATOMIC_RELAXED, __HIP_MEMORY_SCOPE_AGENT);
}

// ---------------------------------------------------------------------------
// Kernel 2: fused GIN MLP  h2 = relu(relu((x+agg)W1+b1)W2+b2), gate = h2.Wg+bg
// 4 waves / block, each wave owns a 16-row tile, FP32 WMMA 16x16x4.
//   A layout (16x4 f32): lane = (k>=2?16:0)+m, vgpr = k%2
//   B layout (4x16 f32): lane = (k>=2?16:0)+n, vgpr = k%2
//   C/D layout (16x16 f32): lane = (m>=8?16:0)+n, vgpr = m%8
// ---------------------------------------------------------------------------
__global__ __launch_bounds__(128) void gin_mlp_kernel(const float* __restrict__ x,
                                                      const float* __restrict__ W1,
                                                      const float* __restrict__ b1,
                                                      const float* __restrict__ W2,
                                                      const float* __restrict__ b2,
                                                      const float* __restrict__ Wg,
                                                      const float* __restrict__ bg,
                                                      float* __restrict__ buf,   // in: agg, out: h2
                                                      float* __restrict__ gate) {
    __shared__ __align__(16) float W1t[FDIM * FDIM];   // transposed: W1t[n*64+k]
    __shared__ __align__(16) float W2t[FDIM * FDIM];
    __shared__ __align__(16) float stg[4][16 * FDIM];  // per-wave staging
    __shared__ float b1s[FDIM], b2s[FDIM], wgs[FDIM];

    const int tid = threadIdx.x;
    for (int i = tid; i < FDIM * FDIM; i += 128) {
        int k = i >> 6, n = i & 63;
        W1t[n * FDIM + k] = W1[i];
        W2t[n * FDIM + k] = W2[i];
    }
    if (tid < FDIM) { b1s[tid] = b1[tid]; b2s[tid] = b2[tid]; wgs[tid] = Wg[tid]; }
    __syncthreads();

    const int wv   = tid >> 5;
    const int lane = tid & 31;
    const int tile = blockIdx.x * 4 + wv;
    if (tile * 16 >= N_NODES) return;     // wave-uniform -> EXEC stays all-ones
    const int row0 = tile * 16;
    const int mA   = lane & 15;           // row (A) / col (B,C,D) index
    const int kh   = lane >> 4;           // K half

    float* s = stg[wv];

    // ---- Layer 1: A = x + agg ----
    v2f a[16];
    {
        const float2* xp = (const float2*)(x   + (size_t)(row0 + mA) * FDIM);
        const float2* gp = (const float2*)(buf + (size_t)(row0 + mA) * FDIM);
#pragma unroll
        for (int t = 0; t < 16; ++t) {
            int c2 = (4 * t + 2 * kh) >> 1;
            float2 xv = xp[c2];
            float2 av = gp[c2];
            a[t][0] = xv.x + av.x;
            a[t][1] = xv.y + av.y;
        }
    }
    v8f acc[4];
#pragma unroll
    for (int nt = 0; nt < 4; ++nt) {
        float bias = b1s[nt * 16 + mA];
#pragma unroll
        for (int r = 0; r < 8; ++r) acc[nt][r] = bias;
#pragma unroll
        for (int t = 0; t < 16; ++t) {
            const float2 w = *(const float2*)&W1t[(nt * 16 + mA) * FDIM + 4 * t + 2 * kh];
            v2f bf; bf[0] = w.x; bf[1] = w.y;
            acc[nt] = __builtin_amdgcn_wmma_f32_16x16x4_f32(
                false, a[t], false, bf, (short)0, acc[nt], false, false);
        }
    }
    // ReLU + restage in row-major so layer 2 can reload in A layout
#pragma unroll
    for (int nt = 0; nt < 4; ++nt)
#pragma unroll
        for (int r = 0; r < 8; ++r)
            s[(r + 8 * kh) * FDIM + nt * 16 + mA] = fmaxf(acc[nt][r], 0.0f);

    // ---- Layer 2 ----
    v2f a2[16];
#pragma unroll
    for (int t = 0; t < 16; ++t) {
        const float2 w = *(const float2*)&s[mA * FDIM + 4 * t + 2 * kh];
        a2[t][0] = w.x; a2[t][1] = w.y;
    }
    v8f acc2[4];
#pragma unroll
    for (int nt = 0; nt < 4; ++nt) {
        float bias = b2s[nt * 16 + mA];
#pragma unroll
        for (int r = 0; r < 8; ++r) acc2[nt][r] = bias;
#pragma unroll
        for (int t = 0; t < 16; ++t) {
            const float2 w = *(const float2*)&W2t[(nt * 16 + mA) * FDIM + 4 * t + 2 * kh];
            v2f bf; bf[0] = w.x; bf[1] = w.y;
            acc2[nt] = __builtin_amdgcn_wmma_f32_16x16x4_f32(
                false, a2[t], false, bf, (short)0, acc2[nt], false, false);
        }
    }
#pragma unroll
    for (int nt = 0; nt < 4; ++nt)
#pragma unroll
        for (int r = 0; r < 8; ++r)
            s[(r + 8 * kh) * FDIM + nt * 16 + mA] = fmaxf(acc2[nt][r], 0.0f);

    // ---- gate = h2 . Wg + bg : lane pair (m, m+16) split the 64 columns ----
    float gsum = 0.0f;
    const int c0 = kh * 32;
#pragma unroll
    for (int c = 0; c < 32; ++c) gsum += s[mA * FDIM + c0 + c] * wgs[c0 + c];
    gsum += __shfl_xor(gsum, 16, 32);
    if (lane < 16) gate[row0 + mA] = gsum + bg[0];

    // ---- write h2 tile (overwrites agg buffer), coalesced float4 ----
#pragma unroll
    for (int j = 0; j < 8; ++j) {
        int id = lane + 32 * j;           // 0..255 float4 slots in 16x64 tile
        int r  = id >> 4;
        int c4 = (id & 15) * 4;
        *(float4*)(buf + (size_t)(row0 + r) * FDIM + c4) = *(const float4*)&s[r * FDIM + c4];
    }
}

// ---------------------------------------------------------------------------
// Kernel 3: per-graph segment softmax + attentional pooling + BN + linear +
// log_softmax. One 256-thread block per graph; batch[] is sorted.
// ---------------------------------------------------------------------------
__global__ __launch_bounds__(256) void pool_kernel(const float* __restrict__ h2,
                                                   const float* __restrict__ gate,
                                                   const long long* __restrict__ batch,
                                                   const float* __restrict__ gamma,
                                                   const float* __restrict__ beta,
                                                   const float* __restrict__ mean,
                                                   const float* __restrict__ var,
                                                   const float* __restrict__ Wlin,
                                                   const float* __restrict__ blin,
                                                   float* __restrict__ out) {
    const int g   = blockIdx.x;
    const int tid = threadIdx.x;

    // lower_bound(batch, g) / lower_bound(batch, g+1)
    int start, endd;
    {
        int lo = 0, hi = N_NODES;
        while (lo < hi) { int mid = (lo + hi) >> 1; if (batch[mid] < (long long)g) lo = mid + 1; else hi = mid; }
        start = lo;
        hi = N_NODES;
        while (lo < hi) { int mid = (lo + hi) >> 1; if (batch[mid] < (long long)(g + 1)) lo = mid + 1; else hi = mid; }
        endd = lo;
    }

    __shared__ float red[256];
    __shared__ float normS[FDIM];

    // segment max of gate
    float lm = -INFINITY;
    for (int i = start + tid; i < endd; i += 256) lm = fmaxf(lm, gate[i]);
    red[tid] = lm; __syncthreads();
    for (int s = 128; s > 0; s >>= 1) {
        if (tid < s) red[tid] = fmaxf(red[tid], red[tid + s]);
        __syncthreads();
    }
    const float gmax = red[0];
    __syncthreads();

    // segment sum of exp
    float ls = 0.0f;
    for (int i = start + tid; i < endd; i += 256) ls += expf(gate[i] - gmax);
    red[tid] = ls; __syncthreads();
    for (int s = 128; s > 0; s >>= 1) {
        if (tid < s) red[tid] += red[tid + s];
        __syncthreads();
    }
    const float denom = red[0];
    __syncthreads();
    const float scale = denom > 0.0f ? 1.0f / denom : 0.0f;

    // weighted pooling: 4 row-groups x 64 columns
    const int col = tid & 63;
    const int grp = tid >> 6;
    float acc = 0.0f;
    for (int i = start + grp; i < endd; i += 4)
        acc += expf(gate[i] - gmax) * h2[(size_t)i * FDIM + col];
    red[grp * 64 + col] = acc;
    __syncthreads();

    if (tid < FDIM) {
        float pooled = (red[tid] + red[64 + tid] + red[128 + tid] + red[192 + tid]) * scale;
        float nrm = (pooled - mean[tid]) * rsqrtf(var[tid] + 1e-5f) * gamma[tid] + beta[tid];
        normS[tid] = nrm;
    }
    __syncthreads();

    if (tid == 0) {
        float l0 = blin[0], l1 = blin[1];
        for (int h = 0; h < FDIM; ++h) {
            l0 += normS[h] * Wlin[h * 2 + 0];
            l1 += normS[h] * Wlin[h * 2 + 1];
        }
        float m = fmaxf(l0, l1);
        float lse = m + logf(expf(l0 - m) + expf(l1 - m));
        out[g * 2 + 0] = l0 - lse;
        out[g * 2 + 1] = l1 - lse;
    }
}

// ---------------------------------------------------------------------------
extern "C" void kernel_launch(void* const* d_in, const int* in_sizes, int n_in,
                              void* d_out, int out_size, void* d_ws, size_t ws_size,
                              hipStream_t stream) {
    const float*     x     = (const float*)d_in[0];
    const long long* ei    = (const long long*)d_in[1];
    const long long* batch = (const long long*)d_in[2];
    const float*     W1    = (const float*)d_in[3];
    const float*     b1    = (const float*)d_in[4];
    const float*     W2    = (const float*)d_in[5];
    const float*     b2    = (const float*)d_in[6];
    const float*     Wg    = (const float*)d_in[7];
    const float*     bg    = (const float*)d_in[8];
    const float*     gam   = (const float*)d_in[9];
    const float*     bet   = (const float*)d_in[10];
    const float*     mean  = (const float*)d_in[11];
    const float*     var   = (const float*)d_in[12];
    const float*     Wlin  = (const float*)d_in[13];
    const float*     blin  = (const float*)d_in[14];
    float* out = (float*)d_out;

    float* buf  = (float*)d_ws;                         // [N,64] agg, then h2
    float* gate = buf + (size_t)N_NODES * FDIM;         // [N]

    hipMemsetAsync(buf, 0, (size_t)N_NODES * FDIM * sizeof(float), stream);

    {   // edge scatter-add
        int total = N_EDGES * 16;
        agg_kernel<<<(total + 255) / 256, 256, 0, stream>>>(x, ei, buf);
    }
    {   // fused MLP (4 wave-tiles of 16 rows per block)
        int tiles  = N_NODES / 16;                      // 6250
        int blocks = (tiles + 3) / 4;
        gin_mlp_kernel<<<blocks, 128, 0, stream>>>(x, W1, b1, W2, b2, Wg, bg, buf, gate);
    }
    {   // pooling + head
        pool_kernel<<<NGRAPH, 256, 0, stream>>>(buf, gate, batch, gam, bet, mean, var,
                                                Wlin, blin, out);
    }
}